// SmoothAPLoss_80530636800492
// MI455X (gfx1250) — compile-verified
//
#include <hip/hip_runtime.h>
#include <math.h>

typedef float v2f __attribute__((ext_vector_type(2)));
typedef float v8f __attribute__((ext_vector_type(8)));

#define SAP_DELTA 0.01f
#define NEG_BIG   (-1.0e30f)
#define BIG_POS   (3.0e38f)   // med3 upper clamp (> any finite diff we produce)
#define LDS_N     16384       // max N supported by LDS staging (64 KB)

// ---------------------------------------------------------------------------
// Kernel 0: zero the positive counter (workspace is poisoned, not zeroed).
// ---------------------------------------------------------------------------
__global__ void sap_init(int* __restrict__ num_pos) {
    *num_pos = 0;
}

// ---------------------------------------------------------------------------
// Kernel 1: per-row prep.
//   p[i]     = softmax(pred[i])[1] = 1 / (1 + exp(x0 - x1))
//   aVals[i] = (label==1) ? DELTA - p[i] : -1e30   (A-matrix rows, positives)
//   bVals[i] = (label==0) ? p[i]         : -1e30   (B-matrix cols, negatives)
// Masked entries are huge-negative so relu(a_i + b_j) == 0 for invalid pairs.
// ---------------------------------------------------------------------------
__global__ void sap_prep(const float* __restrict__ pred,
                         const int*   __restrict__ labels,
                         float* __restrict__ aVals,
                         float* __restrict__ bVals,
                         int*   __restrict__ num_pos,
                         int n) {
    __shared__ int cnt;
    if (threadIdx.x == 0) cnt = 0;
    __syncthreads();

    int i = blockIdx.x * blockDim.x + threadIdx.x;
    int isPos = 0;
    if (i < n) {
        float x0 = pred[2 * i + 0];
        float x1 = pred[2 * i + 1];
        float p  = 1.0f / (1.0f + expf(x0 - x1));
        int lab  = labels[i];
        isPos    = (lab == 1) ? 1 : 0;
        aVals[i] = isPos        ? (SAP_DELTA - p) : NEG_BIG;
        bVals[i] = (lab == 0)   ? p               : NEG_BIG;
    }
    atomicAdd(&cnt, isPos);              // LDS atomic (ds_add)
    __syncthreads();
    if (threadIdx.x == 0) atomicAdd(num_pos, cnt);  // int atomic: deterministic
}

// ---------------------------------------------------------------------------
// Kernel 2: pairwise tiles via V_WMMA_F32_16X16X4_F32.
// diff[i][j] = a_i + b_j is rank-2:  A row i = (a_i, 1, 0, 0),
//                                    B col j = (1, b_j, 0, 0)^T,  C = 0.
// One WMMA emits a full 16x16 diff tile (256 pair terms). relu is a single
// v_med3_f32 per element (clamp to [0, BIG_POS]) — avoids the 2-op
// canonicalize+max sequence fmaxf generates — and accumulation pairs into
// v_pk_add_f32. bVals staged once per block in LDS; each wave owns one
// 16-row tile and sweeps all column tiles (2 per ds_load_2addr_b32).
// ---------------------------------------------------------------------------
__global__ __launch_bounds__(256) void sap_pairs(
        const float* __restrict__ aVals,
        const float* __restrict__ bVals,
        double* __restrict__ blockSums,
        int n) {
    __shared__ float ldsB[LDS_N];        // 64 KB

    const int tid = threadIdx.x;

    // Cooperative vectorized stage of bVals into LDS (n multiple of 4).
    {
        const float4* src = (const float4*)bVals;
        float4*       dst = (float4*)ldsB;
        const int n4 = n >> 2;
        for (int k = tid; k < n4; k += blockDim.x) dst[k] = src[k];
        // pad tail to a multiple of 16 with masked values
        const int nPadT = (n + 15) & ~15;
        for (int k = n + tid; k < nPadT; k += blockDim.x) ldsB[k] = NEG_BIG;
    }
    __syncthreads();

    const int lane   = tid & 31;
    const int wave   = tid >> 5;
    const int tileI  = blockIdx.x * 8 + wave;   // row tile owned by this wave
    const int i0     = tileI * 16;
    const int lane15 = lane & 15;
    const int nPad   = (n + 15) & ~15;

    float s = 0.0f;
    if (i0 < n) {
        // A operand (16x4 f32): lanes 0-15 hold {a_i, 1} (K=0,1),
        // lanes 16-31 hold {0, 0} (K=2,3).
        v2f a;
        if (lane < 16) {
            int idx = i0 + lane;
            a.x = (idx < n) ? aVals[idx] : NEG_BIG;
            a.y = 1.0f;
        } else {
            a.x = 0.0f;
            a.y = 0.0f;
        }

        const float bx = (lane < 16) ? 1.0f : 0.0f;   // loop-invariant B row K=0

        v8f acc = {0.f, 0.f, 0.f, 0.f, 0.f, 0.f, 0.f, 0.f};

        #pragma unroll 4
        for (int j0 = 0; j0 < nPad; j0 += 16) {
            float bv = ldsB[j0 + lane15];
            v2f b;
            b.x = bx;
            b.y = (lane < 16) ? bv : 0.0f;             // B row K=1 = b_j
            v8f c = {0.f, 0.f, 0.f, 0.f, 0.f, 0.f, 0.f, 0.f};
            // v_wmma_f32_16x16x4_f32: D[i][j] = a_i + b_j
            c = __builtin_amdgcn_wmma_f32_16x16x4_f32(
                    false, a, false, b, (short)0, c, false, false);
            #pragma unroll
            for (int k = 0; k < 8; ++k) {
                // relu in one v_med3_f32 (inputs are always finite, no NaN)
                acc[k] += __builtin_amdgcn_fmed3f(c[k], 0.0f, BIG_POS);
            }
        }
        #pragma unroll
        for (int k = 0; k < 8; ++k) s += acc[k];
    }

    // Block reduction (reuse LDS, all waves past the read phase).
    __syncthreads();
    ldsB[tid] = s;
    __syncthreads();
    if (tid == 0) {
        double bs = 0.0;
        for (int t = 0; t < 256; ++t) bs += (double)ldsB[t];
        blockSums[blockIdx.x] = bs;   // fixed slot: deterministic
    }
}

// ---------------------------------------------------------------------------
// Kernel 3: deterministic final reduction + divide by num_pos.
// ---------------------------------------------------------------------------
__global__ void sap_finalize(const double* __restrict__ blockSums,
                             const int* __restrict__ num_pos,
                             float* __restrict__ out,
                             int nBlocks) {
    double t = 0.0;
    for (int b = 0; b < nBlocks; ++b) t += blockSums[b];
    float np = (float)(*num_pos);
    out[0] = (float)(t / (double)fmaxf(np, 1.0f));
}

// ---------------------------------------------------------------------------
// Launch
// ---------------------------------------------------------------------------
extern "C" void kernel_launch(void* const* d_in, const int* in_sizes, int n_in,
                              void* d_out, int out_size, void* d_ws, size_t ws_size,
                              hipStream_t stream) {
    const float* pred   = (const float*)d_in[0];   // [N, 2] f32
    const int*   labels = (const int*)d_in[1];     // [N] int
    float*       out    = (float*)d_out;           // scalar f32

    const int n = in_sizes[1];                     // N = 16384

    // Workspace layout (bytes):
    //   [0..3]            int   num_pos
    //   [16 .. 16+4n)     float aVals[n]
    //   [.. +4n)          float bVals[n]
    //   [.. +8*G)         double blockSums[G]     (8-byte aligned: 16+8n)
    char*   ws        = (char*)d_ws;
    int*    num_pos   = (int*)ws;
    float*  aVals     = (float*)(ws + 16);
    float*  bVals     = aVals + n;
    double* blockSums = (double*)(ws + 16 + (size_t)8 * n);

    const int numTiles = (n + 15) / 16;
    const int grid     = (numTiles + 7) / 8;       // 8 waves (row tiles) / block

    sap_init<<<1, 1, 0, stream>>>(num_pos);
    sap_prep<<<(n + 255) / 256, 256, 0, stream>>>(pred, labels, aVals, bVals,
                                                  num_pos, n);
    sap_pairs<<<grid, 256, 0, stream>>>(aVals, bVals, blockSums, n);
    sap_finalize<<<1, 1, 0, stream>>>(blockSums, num_pos, out, grid);
}